// BCSM_84241488544292
// MI455X (gfx1250) — compile-verified
//
#include <hip/hip_runtime.h>
#include <math.h>

// ---------------- dims ----------------
// DM=256, DI=512, DS=16, DR=16, DC=4, B=16, T=5, bt=80
// 8 chains: c<4 -> audio chain i=c (layers 3i+j), c>=4 -> video chain i=c-4 (layers 12+3i+j)

typedef __attribute__((ext_vector_type(2))) float v2f;
typedef __attribute__((ext_vector_type(8))) float v8f;

// workspace layout (in floats)
constexpr size_t U_OFF   = 0;                          // 8 * 80*256
constexpr size_t F_OFF   = U_OFF   + (size_t)8*80*256; // 163840
constexpr size_t XR_OFF  = F_OFF   + (size_t)8*80*256; // 327680 (pre-conv x)
constexpr size_t X_OFF   = XR_OFF  + (size_t)8*80*512; // 655360 (post conv+silu)
constexpr size_t Z_OFF   = X_OFF   + (size_t)8*80*512; // 983040
constexpr size_t DBL_OFF = Z_OFF   + (size_t)8*80*512; // 1310720 (80x48)
constexpr size_t DT_OFF  = DBL_OFF + (size_t)8*80*48;  // 1341440
constexpr size_t Y_OFF   = DT_OFF  + (size_t)8*80*512; // 1669120
constexpr size_t PL_OFF  = Y_OFF   + (size_t)8*80*512; // 1996800 (4 x 80x256 pooled)
constexpr size_t AG_OFF  = PL_OFF  + (size_t)4*80*256; // 2078720
constexpr size_t VG_OFF  = AG_OFF  + (size_t)4*80*256; // 2160640
// total 2242560 floats ~ 9.0 MB

__device__ __forceinline__ int layer_of(int c, int j) {
    return (c < 4) ? (3 * c + j) : (12 + 3 * (c - 4) + j);
}
__device__ __forceinline__ float sigmoidf_(float v) { return 1.f / (1.f + __expf(-v)); }
__device__ __forceinline__ float siluf_(float v)    { return v * sigmoidf_(v); }
__device__ __forceinline__ float softplusf_(float v){ return (v > 20.f) ? v : log1pf(__expf(v)); }

// ---------------- WMMA f32 16x16x4 tile core ----------------
// A: (M x K) row-major, lda; W: (N x K) row-major (i.e. computes A @ W^T), ldw.
// A-matrix layout: lanes0-15 row M=lane, VGPR0/1 = K 0/1; lanes16-31 = K 2/3.
// B-matrix mirrors (col = lane&15).
__device__ __forceinline__ v8f gemm_tile(const float* __restrict__ A, int lda,
                                         const float* __restrict__ W, int ldw,
                                         int K, int m0, int n0) {
    int lane = threadIdx.x & 31;
    int rw = lane & 15;
    int ko = (lane >> 4) << 1; // 0 or 2
    const float* ar = A + (size_t)(m0 + rw) * lda + ko;
    const float* wr = W + (size_t)(n0 + rw) * ldw + ko;
    v8f acc;
#pragma unroll
    for (int i = 0; i < 8; ++i) acc[i] = 0.f;
    for (int k = 0; k < K; k += 4) {
        v2f a, b;
        a.x = ar[k]; a.y = ar[k + 1];
        b.x = wr[k]; b.y = wr[k + 1];
        acc = __builtin_amdgcn_wmma_f32_16x16x4_f32(false, a, false, b, (short)0, acc,
                                                    false, false);
    }
    return acc;
}

// C/D layout: VGPR v -> row m0+v (lanes 0-15) or m0+8+v (lanes 16-31), col = n0 + (lane&15)
__device__ __forceinline__ void store_tile(v8f acc, float* __restrict__ C, int ldc,
                                           int m0, int n0,
                                           const float* __restrict__ bias,
                                           const float* __restrict__ addm, int act) {
    int lane = threadIdx.x & 31;
    int col = lane & 15;
    int rb = m0 + ((lane >> 4) << 3);
    float bv = bias ? bias[n0 + col] : 0.f;
#pragma unroll
    for (int v = 0; v < 8; ++v) {
        size_t off = (size_t)(rb + v) * ldc + n0 + col;
        float val = acc[v] + bv;
        if (addm) val += addm[off];
        if (act)  val = siluf_(val);
        C[off] = val;
    }
}

// ---------------- kernels ----------------

// mean over HW for one tensor: grid(80,256), block 128
__global__ void k_pool(const float* __restrict__ x, float* __restrict__ pooled, int HW) {
    int r = blockIdx.x, ch = blockIdx.y, t = threadIdx.x;
    const float* p = x + ((size_t)r * 256 + ch) * HW;
    float s = 0.f;
    for (int i = t; i < HW; i += 128) s += p[i];
    __shared__ float red[128];
    red[t] = s; __syncthreads();
    for (int st = 64; st > 0; st >>= 1) { if (t < st) red[t] += red[t + st]; __syncthreads(); }
    if (t == 0) pooled[(size_t)r * 256 + ch] = red[0] / (float)HW;
}

// U[0..3] = audio (B-major 80x256)
__global__ void k_copy_audio(float* __restrict__ ws, const float* __restrict__ audio) {
    int idx = blockIdx.x * 256 + threadIdx.x; // 4*20480
    if (idx >= 4 * 20480) return;
    int c = idx / 20480, rem = idx % 20480;
    ws[U_OFF + (size_t)c * 20480 + rem] = audio[rem];
}

// feats[i] = pooled[i] @ vfc_W[i]^T + vfc_b[i] -> U[4+i]; grid(80,1,4) block 32
__global__ void k_gemm_feats(float* __restrict__ ws, const float* __restrict__ vfc_W,
                             const float* __restrict__ vfc_b) {
    int i = blockIdx.z;
    int mt = blockIdx.x % 5, nt = blockIdx.x / 5;
    const float* A = ws + PL_OFF + (size_t)i * 20480;
    const float* W = vfc_W + (size_t)i * 256 * 256;
    v8f acc = gemm_tile(A, 256, W, 256, 256, mt * 16, nt * 16);
    store_tile(acc, ws + U_OFF + (size_t)(4 + i) * 20480, 256, mt * 16, nt * 16,
               vfc_b + (size_t)i * 256, nullptr, 0);
}

// layernorm: F = LN(U) * g + b ; grid(80,1,8) block 256
__global__ void k_ln(float* __restrict__ ws, const float* __restrict__ g,
                     const float* __restrict__ bta, int level) {
    __shared__ float red[256];
    int c = blockIdx.z, r = blockIdx.x, t = threadIdx.x;
    int L = layer_of(c, level);
    float v = ws[U_OFF + (size_t)c * 20480 + (size_t)r * 256 + t];
    red[t] = v; __syncthreads();
    for (int s = 128; s > 0; s >>= 1) { if (t < s) red[t] += red[t + s]; __syncthreads(); }
    float mu = red[0] * (1.f / 256.f); __syncthreads();
    float d = v - mu;
    red[t] = d * d; __syncthreads();
    for (int s = 128; s > 0; s >>= 1) { if (t < s) red[t] += red[t + s]; __syncthreads(); }
    float var = red[0] * (1.f / 256.f);
    float out = d * rsqrtf(var + 1e-5f) * g[(size_t)L * 256 + t] + bta[(size_t)L * 256 + t];
    ws[F_OFF + (size_t)c * 20480 + (size_t)r * 256 + t] = out;
}

// XZ = F @ in_W^T (80 x 1024, K=256): first 512 -> XRAW, last 512 -> Z; grid(320,1,8) block 32
__global__ void k_gemm_in(float* __restrict__ ws, const float* __restrict__ in_W, int level) {
    int c = blockIdx.z, L = layer_of(c, level);
    int mt = blockIdx.x % 5, nt = blockIdx.x / 5;
    const float* A = ws + F_OFF + (size_t)c * 20480;
    const float* W = in_W + (size_t)L * 1024 * 256;
    v8f acc = gemm_tile(A, 256, W, 256, 256, mt * 16, nt * 16);
    int n0 = nt * 16;
    float* dst = (n0 < 512) ? (ws + XR_OFF + (size_t)c * 40960)
                            : (ws + Z_OFF + (size_t)c * 40960);
    if (n0 >= 512) n0 -= 512;
    store_tile(acc, dst, 512, mt * 16, n0, nullptr, nullptr, 0);
}

// causal depthwise conv (width 4 over time within each batch of 5) + bias + silu
__global__ void k_conv(float* __restrict__ ws, const float* __restrict__ conv_W,
                       const float* __restrict__ conv_b, int level) {
    int idx = blockIdx.x * 256 + threadIdx.x; // 8*80*512
    if (idx >= 8 * 40960) return;
    int c = idx / 40960, rem = idx % 40960;
    int r = rem / 512, d = rem % 512;
    int L = layer_of(c, level);
    int b = r / 5, t = r % 5;
    const float* xr = ws + XR_OFF + (size_t)c * 40960;
    const float* w = conv_W + (size_t)L * 2048 + (size_t)d * 4;
    float s = 0.f;
#pragma unroll
    for (int k = 0; k < 4; ++k) {
        int ti = t - 3 + k;
        if (ti >= 0) s += w[k] * xr[(size_t)(b * 5 + ti) * 512 + d];
    }
    s += conv_b[(size_t)L * 512 + d];
    ws[X_OFF + (size_t)c * 40960 + (size_t)r * 512 + d] = siluf_(s);
}

// DBL = X @ xproj_W^T (80 x 48, K=512); grid(15,1,8) block 32
__global__ void k_gemm_xproj(float* __restrict__ ws, const float* __restrict__ xproj_W, int level) {
    int c = blockIdx.z, L = layer_of(c, level);
    int mt = blockIdx.x % 5, nt = blockIdx.x / 5;
    const float* A = ws + X_OFF + (size_t)c * 40960;
    const float* W = xproj_W + (size_t)L * 48 * 512;
    v8f acc = gemm_tile(A, 512, W, 512, 512, mt * 16, nt * 16);
    store_tile(acc, ws + DBL_OFF + (size_t)c * 3840, 48, mt * 16, nt * 16, nullptr, nullptr, 0);
}

// DT = softplus(DBL[:, :16] @ dt_W^T + dt_b)  (80 x 512, K=16)
__global__ void k_dt(float* __restrict__ ws, const float* __restrict__ dt_W,
                     const float* __restrict__ dt_b, int level) {
    int idx = blockIdx.x * 256 + threadIdx.x; // 8*80*512
    if (idx >= 8 * 40960) return;
    int c = idx / 40960, rem = idx % 40960;
    int r = rem / 512, d = rem % 512;
    int L = layer_of(c, level);
    const float* dbl = ws + DBL_OFF + (size_t)c * 3840 + (size_t)r * 48;
    const float* w = dt_W + (size_t)L * 512 * 16 + (size_t)d * 16;
    float s = dt_b[(size_t)L * 512 + d];
#pragma unroll
    for (int j = 0; j < 16; ++j) s += dbl[j] * w[j];
    ws[DT_OFF + (size_t)c * 40960 + (size_t)r * 512 + d] = softplusf_(s);
}

// selective scan: one thread per (chain, b, d); 16 states in registers, 5 time steps
__global__ void k_scan(float* __restrict__ ws, const float* __restrict__ A_log,
                       const float* __restrict__ Dp, int level) {
    int idx = blockIdx.x * 256 + threadIdx.x; // 8*16*512 = 65536
    if (idx >= 8 * 8192) return;
    int c = idx / 8192, rem = idx % 8192;
    int b = rem / 512, d = rem % 512;
    int L = layer_of(c, level);
    float Acoef[16];
#pragma unroll
    for (int n = 0; n < 16; ++n)
        Acoef[n] = -__expf(A_log[(size_t)L * 8192 + (size_t)d * 16 + n]);
    float h[16];
#pragma unroll
    for (int n = 0; n < 16; ++n) h[n] = 0.f;
    float Dv = Dp[(size_t)L * 512 + d];
    for (int t = 0; t < 5; ++t) {
        int r = b * 5 + t;
        float dt = ws[DT_OFF + (size_t)c * 40960 + (size_t)r * 512 + d];
        float xv = ws[X_OFF  + (size_t)c * 40960 + (size_t)r * 512 + d];
        const float* dbl = ws + DBL_OFF + (size_t)c * 3840 + (size_t)r * 48;
        float y = 0.f;
#pragma unroll
        for (int n = 0; n < 16; ++n) {
            float dA = __expf(dt * Acoef[n]);
            h[n] = dA * h[n] + dt * dbl[16 + n] * xv;
            y += h[n] * dbl[32 + n];
        }
        y += xv * Dv;
        float z = ws[Z_OFF + (size_t)c * 40960 + (size_t)r * 512 + d];
        y *= siluf_(z);
        ws[Y_OFF + (size_t)c * 40960 + (size_t)r * 512 + d] = y;
    }
}

// U = F + Y @ out_W^T (80 x 256, K=512); grid(80,1,8) block 32
__global__ void k_gemm_out(float* __restrict__ ws, const float* __restrict__ out_W, int level) {
    int c = blockIdx.z, L = layer_of(c, level);
    int mt = blockIdx.x % 5, nt = blockIdx.x / 5;
    const float* A = ws + Y_OFF + (size_t)c * 40960;
    const float* W = out_W + (size_t)L * 256 * 512;
    v8f acc = gemm_tile(A, 512, W, 512, 512, mt * 16, nt * 16);
    store_tile(acc, ws + U_OFF + (size_t)c * 20480, 256, mt * 16, nt * 16,
               nullptr, ws + F_OFF + (size_t)c * 20480, 0);
}

// gates: silu(U[c] @ {ag|vg}_W^T + b) -> AG/VG; grid(80,1,8) block 32
__global__ void k_gemm_gate(float* __restrict__ ws, const float* __restrict__ ag_W,
                            const float* __restrict__ ag_b, const float* __restrict__ vg_W,
                            const float* __restrict__ vg_b) {
    int c = blockIdx.z;
    int mt = blockIdx.x % 5, nt = blockIdx.x / 5;
    const float* A = ws + U_OFF + (size_t)c * 20480;
    const float* W; const float* bias; float* C;
    if (c < 4) { W = ag_W + (size_t)c * 65536; bias = ag_b + (size_t)c * 256;
                 C = ws + AG_OFF + (size_t)c * 20480; }
    else       { W = vg_W + (size_t)(c - 4) * 65536; bias = vg_b + (size_t)(c - 4) * 256;
                 C = ws + VG_OFF + (size_t)(c - 4) * 20480; }
    v8f acc = gemm_tile(A, 256, W, 256, 256, mt * 16, nt * 16);
    store_tile(acc, C, 256, mt * 16, nt * 16, bias, nullptr, 1);
}

// out = x * (1 + gate[row]) vectorized float4 (requires HW % 4 == 0)
__global__ void k_apply4(const float* __restrict__ x, float* __restrict__ out,
                         const float* __restrict__ gate, int HW, int trans, long n4) {
    long i = (long)blockIdx.x * 256 + threadIdx.x;
    if (i >= n4) return;
    long e = i * 4;
    long rc = e / HW;
    int ch = (int)(rc % 256);
    int r = (int)(rc / 256);
    int gr = trans ? (r % 16) * 5 + r / 16 : r;
    float gv = 1.f + gate[(size_t)gr * 256 + ch];
    float4 v = ((const float4*)x)[i];
    v.x *= gv; v.y *= gv; v.z *= gv; v.w *= gv;
    ((float4*)out)[i] = v;
}

// scalar version for HW % 4 != 0 (x4, HW=49)
__global__ void k_apply1(const float* __restrict__ x, float* __restrict__ out,
                         const float* __restrict__ gate, int HW, int trans, long n) {
    long i = (long)blockIdx.x * 256 + threadIdx.x;
    if (i >= n) return;
    long rc = i / HW;
    int ch = (int)(rc % 256);
    int r = (int)(rc / 256);
    int gr = trans ? (r % 16) * 5 + r / 16 : r;
    out[i] = x[i] * (1.f + gate[(size_t)gr * 256 + ch]);
}

// audio_out = audio * (1 + mean of 4 transposed video gates)
__global__ void k_audio_out(const float* __restrict__ audio, float* __restrict__ out,
                            const float* __restrict__ ws) {
    int idx = blockIdx.x * 256 + threadIdx.x; // 80*256
    if (idx >= 20480) return;
    int r = idx / 256, ch = idx % 256;
    int tr = (r % 16) * 5 + r / 16;
    float g = 0.f;
#pragma unroll
    for (int i = 0; i < 4; ++i) g += ws[VG_OFF + (size_t)i * 20480 + (size_t)tr * 256 + ch];
    g *= 0.25f;
    out[idx] = audio[idx] * (1.f + g);
}

extern "C" void kernel_launch(void* const* d_in, const int* in_sizes, int n_in,
                              void* d_out, int out_size, void* d_ws, size_t ws_size,
                              hipStream_t stream) {
    const float* x1     = (const float*)d_in[0];
    const float* x2     = (const float*)d_in[1];
    const float* x3     = (const float*)d_in[2];
    const float* x4     = (const float*)d_in[3];
    const float* audio  = (const float*)d_in[4];
    const float* vfc_W  = (const float*)d_in[5];
    const float* vfc_b  = (const float*)d_in[6];
    const float* ln_g   = (const float*)d_in[7];
    const float* ln_b   = (const float*)d_in[8];
    const float* in_W   = (const float*)d_in[9];
    const float* conv_W = (const float*)d_in[10];
    const float* conv_b = (const float*)d_in[11];
    const float* xprj_W = (const float*)d_in[12];
    const float* dt_W   = (const float*)d_in[13];
    const float* dt_b   = (const float*)d_in[14];
    const float* A_log  = (const float*)d_in[15];
    const float* Dp     = (const float*)d_in[16];
    const float* out_W  = (const float*)d_in[17];
    const float* ag_W   = (const float*)d_in[18];
    const float* ag_b   = (const float*)d_in[19];
    const float* vg_W   = (const float*)d_in[20];
    const float* vg_b   = (const float*)d_in[21];
    float* out = (float*)d_out;
    float* ws = (float*)d_ws;

    // --- pooling: x_i -> pooled[i] (80x256 each) ---
    k_pool<<<dim3(80, 256), 128, 0, stream>>>(x1, ws + PL_OFF + 0 * 20480, 56 * 56);
    k_pool<<<dim3(80, 256), 128, 0, stream>>>(x2, ws + PL_OFF + 1 * 20480, 28 * 28);
    k_pool<<<dim3(80, 256), 128, 0, stream>>>(x3, ws + PL_OFF + 2 * 20480, 14 * 14);
    k_pool<<<dim3(80, 256), 128, 0, stream>>>(x4, ws + PL_OFF + 3 * 20480, 7 * 7);

    // --- chain inputs ---
    k_copy_audio<<<320, 256, 0, stream>>>(ws, audio);
    k_gemm_feats<<<dim3(80, 1, 4), 32, 0, stream>>>(ws, vfc_W, vfc_b);

    // --- 3 levels of 8 concurrent mamba blocks ---
    for (int j = 0; j < 3; ++j) {
        k_ln<<<dim3(80, 1, 8), 256, 0, stream>>>(ws, ln_g, ln_b, j);
        k_gemm_in<<<dim3(320, 1, 8), 32, 0, stream>>>(ws, in_W, j);
        k_conv<<<1280, 256, 0, stream>>>(ws, conv_W, conv_b, j);
        k_gemm_xproj<<<dim3(15, 1, 8), 32, 0, stream>>>(ws, xprj_W, j);
        k_dt<<<1280, 256, 0, stream>>>(ws, dt_W, dt_b, j);
        k_scan<<<256, 256, 0, stream>>>(ws, A_log, Dp, j);
        k_gemm_out<<<dim3(80, 1, 8), 32, 0, stream>>>(ws, out_W, j);
    }

    // --- gates ---
    k_gemm_gate<<<dim3(80, 1, 8), 32, 0, stream>>>(ws, ag_W, ag_b, vg_W, vg_b);

    // --- elementwise outputs: order (x4o, x3o, x2o, x1o, audio_out) ---
    const long n_x4 = 80L * 256 * 49;
    const long n_x3 = 80L * 256 * 196;
    const long n_x2 = 80L * 256 * 784;
    const long n_x1 = 80L * 256 * 3136;
    float* o_x4 = out;
    float* o_x3 = o_x4 + n_x4;
    float* o_x2 = o_x3 + n_x3;
    float* o_x1 = o_x2 + n_x2;
    float* o_au = o_x1 + n_x1;

    // x4: gate = transposed ag[3]; HW=49 scalar
    k_apply1<<<(int)((n_x4 + 255) / 256), 256, 0, stream>>>(
        x4, o_x4, ws + AG_OFF + 3 * 20480, 49, 1, n_x4);
    // x3: transposed ag[2]
    k_apply4<<<(int)((n_x3 / 4 + 255) / 256), 256, 0, stream>>>(
        x3, o_x3, ws + AG_OFF + 2 * 20480, 196, 1, n_x3 / 4);
    // x2: transposed ag[1]
    k_apply4<<<(int)((n_x2 / 4 + 255) / 256), 256, 0, stream>>>(
        x2, o_x2, ws + AG_OFF + 1 * 20480, 784, 1, n_x2 / 4);
    // x1: non-transposed ag[0]
    k_apply4<<<(int)((n_x1 / 4 + 255) / 256), 256, 0, stream>>>(
        x1, o_x1, ws + AG_OFF + 0 * 20480, 3136, 0, n_x1 / 4);

    k_audio_out<<<80, 256, 0, stream>>>(audio, o_au, ws);

    (void)in_sizes; (void)n_in; (void)out_size; (void)ws_size;
}